// TemporalGCN_85409719648313
// MI455X (gfx1250) — compile-verified
//
#include <hip/hip_runtime.h>
#include <hip/hip_bf16.h>

typedef __attribute__((ext_vector_type(2))) float v2f;
typedef __attribute__((ext_vector_type(8))) float v8f;

#define WMMA_F32(a, b, c) \
    __builtin_amdgcn_wmma_f32_16x16x4_f32(false, (a), false, (b), (short)0, (c), false, false)

__device__ inline void atomAddF(float* p, float v) {
#if defined(__HIP_DEVICE_COMPILE__)
    unsafeAtomicAdd(p, v);   // hardware global_atomic_add_f32 (no CAS loop)
#else
    atomicAdd(p, v);
#endif
}

// ---------------------------------------------------------------------------
// degree + normalization
// ---------------------------------------------------------------------------
__global__ void deg_count(const int* __restrict__ col, unsigned* __restrict__ deg, int E) {
    int e = blockIdx.x * 256 + threadIdx.x;
    if (e < E) atomicAdd(&deg[col[e]], 1u);
}

__global__ void make_dis(const unsigned* __restrict__ deg, float* __restrict__ dis, int N) {
    int n = blockIdx.x * 256 + threadIdx.x;
    if (n < N) dis[n] = rsqrtf((float)deg[n] + 1.0f);
}

// ---------------------------------------------------------------------------
// Node encoder: x_enc = relu(x @ W_node + b_node)   [16-node tile per wave]
// then xw_z = x_enc @ Wg_z ; xw_h = x_enc @ Wg_h    (via LDS transpose)
// All matmuls on v_wmma_f32_16x16x4_f32.
// ---------------------------------------------------------------------------
__global__ __launch_bounds__(32) void node_encode(
    const float* __restrict__ x, const float* __restrict__ Wn, const float* __restrict__ bn,
    const float* __restrict__ Wgz, const float* __restrict__ Wgh,
    float* __restrict__ xwz, float* __restrict__ xwh)
{
    __shared__ float lds[16 * 36];               // 16 rows x 32 cols, stride 36 (bank-conflict free)
    const int lane = threadIdx.x;
    const int m    = lane & 15;                  // A row / B col on lanes
    const int n    = lane & 15;
    const int koff = (lane >> 4) << 1;           // lanes 16-31 hold K+2,K+3
    const int node0 = blockIdx.x * 16;

    // ---- stage 1: x (16x128) @ W_node (128x32) ----
    v8f c0 = {}, c1 = {};
    const float* xrow = x + (size_t)(node0 + m) * 128;
#pragma unroll
    for (int k = 0; k < 128; k += 4) {
        v2f a = *(const v2f*)(xrow + k + koff);
        v2f b0, b1;
        b0.x = Wn[(k + koff) * 32 + n];      b0.y = Wn[(k + koff + 1) * 32 + n];
        b1.x = Wn[(k + koff) * 32 + 16 + n]; b1.y = Wn[(k + koff + 1) * 32 + 16 + n];
        c0 = WMMA_F32(a, b0, c0);
        c1 = WMMA_F32(a, b1, c1);
    }
    // bias + relu, transpose through LDS (C layout -> A layout)
    const float bb0 = bn[n], bb1 = bn[16 + n];
    const int mrow = (lane >> 4) * 8;
#pragma unroll
    for (int i = 0; i < 8; ++i) {
        lds[(mrow + i) * 36 + n]      = fmaxf(c0[i] + bb0, 0.f);
        lds[(mrow + i) * 36 + 16 + n] = fmaxf(c1[i] + bb1, 0.f);
    }
    __syncthreads();   // single-wave WG: just forces the dscnt wait before re-read

    // ---- stage 2: x_enc (16x32) @ Wg_z and @ Wg_h ----
    v8f z0 = {}, z1 = {}, h0 = {}, h1 = {};
#pragma unroll
    for (int k = 0; k < 32; k += 4) {
        v2f a = *(const v2f*)&lds[m * 36 + k + koff];
        v2f bz0, bz1, bh0, bh1;
        bz0.x = Wgz[(k + koff) * 32 + n];      bz0.y = Wgz[(k + koff + 1) * 32 + n];
        bz1.x = Wgz[(k + koff) * 32 + 16 + n]; bz1.y = Wgz[(k + koff + 1) * 32 + 16 + n];
        bh0.x = Wgh[(k + koff) * 32 + n];      bh0.y = Wgh[(k + koff + 1) * 32 + n];
        bh1.x = Wgh[(k + koff) * 32 + 16 + n]; bh1.y = Wgh[(k + koff + 1) * 32 + 16 + n];
        z0 = WMMA_F32(a, bz0, z0);
        z1 = WMMA_F32(a, bz1, z1);
        h0 = WMMA_F32(a, bh0, h0);
        h1 = WMMA_F32(a, bh1, h1);
    }
#pragma unroll
    for (int i = 0; i < 8; ++i) {
        size_t r = (size_t)(node0 + mrow + i) * 32;
        xwz[r + n] = z0[i]; xwz[r + 16 + n] = z1[i];
        xwh[r + n] = h0[i]; xwh[r + 16 + n] = h1[i];
    }
}

// scalar fallback for a non-multiple-of-16 node tail (not hit for N=100000)
__global__ void node_tail(const float* x, const float* Wn, const float* bn,
                          const float* Wgz, const float* Wgh,
                          float* xwz, float* xwh, int start, int N) {
    int nd = start + blockIdx.x;
    if (nd >= N) return;
    int j = threadIdx.x;
    float acc = bn[j];
    for (int k = 0; k < 128; ++k) acc = fmaf(x[(size_t)nd * 128 + k], Wn[k * 32 + j], acc);
    float xe = fmaxf(acc, 0.f);
    float az = 0.f, ah = 0.f;
    for (int k = 0; k < 32; ++k) {
        float xk = __shfl(xe, k, 32);
        az = fmaf(xk, Wgz[k * 32 + j], az);
        ah = fmaf(xk, Wgh[k * 32 + j], ah);
    }
    xwz[(size_t)nd * 32 + j] = az;
    xwh[(size_t)nd * 32 + j] = ah;
}

// ---------------------------------------------------------------------------
// GCN message scatter: agg[col] += dis[row]*dis[col] * xw[row]  (z and h gates)
// One 32-lane group per edge; arrays are L2-resident (12.8 MB each).
// ---------------------------------------------------------------------------
__global__ __launch_bounds__(256) void scatter_msgs(
    const int* __restrict__ row, const int* __restrict__ col, const float* __restrict__ dis,
    const float* __restrict__ xwz, const float* __restrict__ xwh,
    float* __restrict__ aggz, float* __restrict__ aggh, int E)
{
    int e = blockIdx.x * 8 + (threadIdx.x >> 5);
    int j = threadIdx.x & 31;
    if (e >= E) return;
    int r = row[e], c = col[e];
    float w = dis[r] * dis[c];
    atomAddF(&aggz[(size_t)c * 32 + j], w * xwz[(size_t)r * 32 + j]);
    atomAddF(&aggh[(size_t)c * 32 + j], w * xwh[(size_t)r * 32 + j]);
}

// ---------------------------------------------------------------------------
// Gates: Z = sigm((agg_z + d^2*xw_z + bg_z) @ Wl_z[0:32] + bl_z)
//        Ht = tanh((agg_h + d^2*xw_h + bg_h) @ Wl_h[0:32] + bl_h)
//        h = (1-Z)*Ht ;  a = h.w1 ; b = h.w2           (R branch is exactly 0)
// ---------------------------------------------------------------------------
__global__ __launch_bounds__(256) void gates(
    const float* __restrict__ aggz, const float* __restrict__ aggh,
    const float* __restrict__ xwz,  const float* __restrict__ xwh,
    const float* __restrict__ dis,  const float* __restrict__ bgz, const float* __restrict__ bgh,
    const float* __restrict__ Wlz,  const float* __restrict__ blz,
    const float* __restrict__ Wlh,  const float* __restrict__ blh,
    const float* __restrict__ Wout, float* __restrict__ a_node, float* __restrict__ b_node, int N)
{
    int lane = threadIdx.x & 31;
    int nd = blockIdx.x * 8 + (threadIdx.x >> 5);
    if (nd >= N) return;
    float d = dis[nd], d2 = d * d;
    size_t base = (size_t)nd * 32;
    float vz = aggz[base + lane] + d2 * xwz[base + lane] + bgz[lane];
    float vh = aggh[base + lane] + d2 * xwh[base + lane] + bgh[lane];
    float zacc = blz[lane], hacc = blh[lane];
#pragma unroll
    for (int j = 0; j < 32; ++j) {
        float vzj = __shfl(vz, j, 32);
        float vhj = __shfl(vh, j, 32);
        zacc = fmaf(vzj, Wlz[j * 32 + lane], zacc);
        hacc = fmaf(vhj, Wlh[j * 32 + lane], hacc);
    }
    float Z = 1.f / (1.f + expf(-zacc));
    float T = tanhf(hacc);
    float h = (1.f - Z) * T;
    float av = h * Wout[lane];
    float bv = h * Wout[32 + lane];
#pragma unroll
    for (int s = 1; s < 32; s <<= 1) {
        av += __shfl_xor(av, s, 32);
        bv += __shfl_xor(bv, s, 32);
    }
    if (lane == 0) { a_node[nd] = av; b_node[nd] = bv; }
}

// ---------------------------------------------------------------------------
// Edge head (the big one): 16 edges per wave per tile, grid-stride over tiles.
// eterm = relu(edge_attr @ W_edge + b_edge) . w3  via WMMA f32 + shfl reduce,
// out[e] = a[row] + b[col] + eterm + b_out
// ---------------------------------------------------------------------------
__global__ __launch_bounds__(256) void edge_out(
    const float* __restrict__ ea, const int* __restrict__ row, const int* __restrict__ col,
    const float* __restrict__ We, const float* __restrict__ be,
    const float* __restrict__ a_node, const float* __restrict__ b_node,
    const float* __restrict__ w3, const float* __restrict__ bout,
    float* __restrict__ out, int ntiles)
{
    const int lane = threadIdx.x & 31;
    const int m = lane & 15, n = lane & 15;
    const int koff = (lane >> 4) << 1;
    const int waveId = blockIdx.x * (blockDim.x >> 5) + (threadIdx.x >> 5);
    const int nwaves = gridDim.x * (blockDim.x >> 5);

    // Preload W_edge (32x32) in B layout; amortized over all tiles of this wave.
    v2f wb0[8], wb1[8];
#pragma unroll
    for (int ks = 0; ks < 8; ++ks) {
        int k = ks * 4 + koff;
        wb0[ks].x = We[k * 32 + n];      wb0[ks].y = We[(k + 1) * 32 + n];
        wb1[ks].x = We[k * 32 + 16 + n]; wb1[ks].y = We[(k + 1) * 32 + 16 + n];
    }
    const float be0 = be[n], be1 = be[16 + n];
    const float w30 = w3[n], w31 = w3[16 + n];
    const float bo = bout[0];

    for (int t = waveId; t < ntiles; t += nwaves) {
        const size_t e0 = (size_t)t * 16;
        const float* ar = ea + (e0 + m) * 32;   // edge rows are 128B-aligned cachelines
        v8f c0 = {}, c1 = {};
#pragma unroll
        for (int ks = 0; ks < 8; ++ks) {
            v2f a = *(const v2f*)(ar + ks * 4 + koff);
            c0 = WMMA_F32(a, wb0[ks], c0);
            c1 = WMMA_F32(a, wb1[ks], c1);
        }
        // relu + scale by w3, reduce each C row over N within 16-lane halves
        float u[8];
#pragma unroll
        for (int i = 0; i < 8; ++i) {
            float v = fmaxf(c0[i] + be0, 0.f) * w30 + fmaxf(c1[i] + be1, 0.f) * w31;
            v += __shfl_xor(v, 1, 16);
            v += __shfl_xor(v, 2, 16);
            v += __shfl_xor(v, 4, 16);
            v += __shfl_xor(v, 8, 16);
            u[i] = v;   // lanes 0-15: row i ; lanes 16-31: row 8+i
        }
        int idx = lane & 7;
        float myu = u[0];
#pragma unroll
        for (int i = 1; i < 8; ++i) myu = (idx == i) ? u[i] : myu;
        if ((lane & 15) < 8) {                     // 16 writer lanes, one edge each
            int e = (int)e0 + ((lane >> 4) << 3) + idx;
            out[e] = myu + a_node[row[e]] + b_node[col[e]] + bo;
        }
    }
}

// scalar fallback for a non-multiple-of-16 edge tail (not hit for E=1.6M)
__global__ void edge_tail(const float* ea, const int* row, const int* col,
                          const float* We, const float* be, const float* w3,
                          const float* a_node, const float* b_node, const float* bout,
                          float* out, int start, int E) {
    int e = start + blockIdx.x;
    if (e >= E) return;
    int j = threadIdx.x;
    float acc = be[j];
    for (int k = 0; k < 32; ++k) acc = fmaf(ea[(size_t)e * 32 + k], We[k * 32 + j], acc);
    float v = fmaxf(acc, 0.f) * w3[j];
    for (int s = 1; s < 32; s <<= 1) v += __shfl_xor(v, s, 32);
    if (j == 0) out[e] = v + a_node[row[e]] + b_node[col[e]] + bout[0];
}

// ---------------------------------------------------------------------------
extern "C" void kernel_launch(void* const* d_in, const int* in_sizes, int n_in,
                              void* d_out, int out_size, void* d_ws, size_t ws_size,
                              hipStream_t stream) {
    const float* x    = (const float*)d_in[0];
    const int*   ei   = (const int*)d_in[1];
    const float* ea   = (const float*)d_in[2];
    const float* Wn   = (const float*)d_in[3];
    const float* bn   = (const float*)d_in[4];
    const float* We   = (const float*)d_in[5];
    const float* be   = (const float*)d_in[6];
    const float* Wgz  = (const float*)d_in[7];
    const float* bgz  = (const float*)d_in[8];
    // d_in[9],[10] = Wg_r/bg_r ; d_in[15],[16] = Wl_r/bl_r : provably unused (h0==0)
    const float* Wgh  = (const float*)d_in[11];
    const float* bgh  = (const float*)d_in[12];
    const float* Wlz  = (const float*)d_in[13];
    const float* blz  = (const float*)d_in[14];
    const float* Wlh  = (const float*)d_in[17];
    const float* blh  = (const float*)d_in[18];
    const float* Wout = (const float*)d_in[19];
    const float* bout = (const float*)d_in[20];
    float* out = (float*)d_out;

    const int N = in_sizes[0] / 128;   // F_NODE = 128
    const int E = in_sizes[2] / 32;    // F_EDGE = 32
    const int* row = ei;
    const int* col = ei + E;

    // workspace layout (floats), ~52.6 MB total
    const size_t N32 = (size_t)N * 32;
    float* xwz    = (float*)d_ws;
    float* xwh    = xwz + N32;
    float* aggz   = xwh + N32;
    float* aggh   = aggz + N32;
    float* dis    = aggh + N32;
    float* a_node = dis + N;
    float* b_node = a_node + N;
    unsigned* deg = (unsigned*)(b_node + N);

    hipMemsetAsync(aggz, 0, 2 * N32 * sizeof(float), stream);
    hipMemsetAsync(deg, 0, (size_t)N * sizeof(unsigned), stream);

    deg_count<<<(E + 255) / 256, 256, 0, stream>>>(col, deg, E);
    make_dis<<<(N + 255) / 256, 256, 0, stream>>>(deg, dis, N);

    const int ntilesN = N / 16;
    if (ntilesN)
        node_encode<<<ntilesN, 32, 0, stream>>>(x, Wn, bn, Wgz, Wgh, xwz, xwh);
    const int tailN = N - ntilesN * 16;
    if (tailN)
        node_tail<<<tailN, 32, 0, stream>>>(x, Wn, bn, Wgz, Wgh, xwz, xwh, ntilesN * 16, N);

    scatter_msgs<<<(E + 7) / 8, 256, 0, stream>>>(row, col, dis, xwz, xwh, aggz, aggh, E);

    gates<<<(N + 7) / 8, 256, 0, stream>>>(aggz, aggh, xwz, xwh, dis, bgz, bgh,
                                           Wlz, blz, Wlh, blh, Wout, a_node, b_node, N);

    const int ntilesE = E / 16;
    if (ntilesE)
        edge_out<<<512, 256, 0, stream>>>(ea, row, col, We, be, a_node, b_node,
                                          Wout + 64, bout, out, ntilesE);
    const int tailE = E - ntilesE * 16;
    if (tailE)
        edge_tail<<<tailE, 32, 0, stream>>>(ea, row, col, We, be, Wout + 64,
                                            a_node, b_node, bout, out, ntilesE * 16, E);
}